// FunctionalMapNet_46411416601226
// MI455X (gfx1250) — compile-verified
//
#include <hip/hip_runtime.h>

typedef __attribute__((ext_vector_type(2))) float v2f;
typedef __attribute__((ext_vector_type(8))) float v8f;

#define B_ 32
#define V_ 8192
#define F_ 256
#define K_ 128

// Async global->LDS copy, 16B per lane, tracked by ASYNCcnt (CDNA5 path).
__device__ __forceinline__ void async_b128(unsigned lds_off, const float* gptr) {
    asm volatile("global_load_async_to_lds_b128 %0, %1, off"
                 :: "v"(lds_off), "v"((unsigned long long)(uintptr_t)gptr)
                 : "memory");
}
__device__ __forceinline__ void wait_async0() {
    asm volatile("s_wait_asynccnt 0x0" ::: "memory");
}

// ---------------------------------------------------------------------------
// Kernel 1: spectral projection  C[b,m,n] = sum_v A[b,m,v] * X[b,v,n]
//   A = evecs_trans [B,K,V], X = feat [B,V,F], C = f_hat/g_hat [B,K,F]
// WG tile 64(M) x 128(N), BK=32, double-buffered ASYNC global->LDS staging.
// 8 waves in a 2x4 grid; each wave owns a 2x2 grid of 16x16 WMMA subtiles
// (every A/B fragment feeds two V_WMMA_F32_16X16X4_F32).
// blockIdx.z selects the (evecs_x,feat_x)->f_hat or (y)->g_hat problem.
// ---------------------------------------------------------------------------
__global__ __launch_bounds__(256) void proj_gemm(
    const float* __restrict__ evx, const float* __restrict__ fx,
    const float* __restrict__ evy, const float* __restrict__ fy,
    float* __restrict__ f_hat, float* __restrict__ g_hat)
{
    // A row-major [m][v]: stride 36 floats (144B: B128-aligned, 16-bank clean)
    // B row-major [v][n]: stride 136 floats (544B: B128-aligned; rows k,k+2
    //                     land on disjoint 16-bank sets for the half-waves)
    __shared__ float As[2][64][36];
    __shared__ float Bs[2][32][136];
    constexpr unsigned ABUF = 64u * 36u * 4u;     // one A buffer, bytes
    constexpr unsigned BBUF = 32u * 136u * 4u;    // one B buffer, bytes

    const int b    = blockIdx.y;
    const int tm   = (blockIdx.x >> 1) * 64;      // K tile (2)
    const int tn   = (blockIdx.x & 1)  * 128;     // F tile (2)
    const int t    = threadIdx.x;
    const int lane = t & 31;
    const int wave = t >> 5;
    const int l16  = lane & 15;
    const int khalf = (lane >> 4) << 1;           // lanes 16-31: K+2,K+3
    const int mhalf = (lane >> 4) * 8;            // C/D: lanes 16-31: M+8

    // wave -> 32x32 region: 2x4 wave grid over 64x128
    const int m0 = (wave >> 2) * 32, m1 = m0 + 16;
    const int n0 = (wave & 3)  * 32, n1 = n0 + 16;

    const float* Ag = (blockIdx.z ? evy : evx) + (size_t)b * K_ * V_;
    const float* Xg = (blockIdx.z ? fy  : fx ) + (size_t)b * V_ * F_;
    float*       Cg = (blockIdx.z ? g_hat : f_hat) + (size_t)b * K_ * F_;

    // Per-thread async-copy coordinates (16B segments).
    // A tile 64x32: 512 segs -> 2 per thread (rows mi, mi+32)
    const int miA   = t >> 3;
    const int vsegA = (t & 7) * 4;
    const float*   gA   = Ag + (size_t)(tm + miA) * V_ + vsegA;
    const unsigned aoff = (unsigned)(uintptr_t)&As[0][miA][vsegA];
    // B tile 32x128: 1024 segs -> 4 per thread (rows viB + 8r)
    const int viB   = t >> 5;
    const int nsegB = (t & 31) * 4;
    const float*   gB   = Xg + (size_t)viB * F_ + tn + nsegB;
    const unsigned boff = (unsigned)(uintptr_t)&Bs[0][viB][nsegB];

    auto issue = [&](int chunk) {
        const int v0 = chunk * 32;
        const unsigned ab = (chunk & 1) ? ABUF : 0u;
        const unsigned bb = (chunk & 1) ? BBUF : 0u;
        async_b128(aoff + ab,               gA + v0);
        async_b128(aoff + ab + 32u*36u*4u,  gA + v0 + (size_t)32 * V_);
        #pragma unroll
        for (int r = 0; r < 4; ++r)
            async_b128(boff + bb + (unsigned)r * (8u*136u*4u),
                       gB + (size_t)v0 * F_ + (size_t)r * 8 * F_);
    };

    v8f acc00 = {}, acc01 = {}, acc10 = {}, acc11 = {};

    const int nch = V_ / 32;
    issue(0);
    for (int i = 0; i < nch; ++i) {
        wait_async0();            // this wave's chunk-i writes landed in LDS
        __syncthreads();          // all waves' writes landed; prev readers done
        if (i + 1 < nch) issue(i + 1);   // overlap next chunk with compute

        const float (*Asb)[36]  = As[i & 1];
        const float (*Bsb)[136] = Bs[i & 1];
        #pragma unroll
        for (int kk = 0; kk < 32; kk += 4) {
            const int kidx = kk + khalf;
            v2f am0, am1, bn0, bn1;
            am0.x = Asb[m0 + l16][kidx]; am0.y = Asb[m0 + l16][kidx + 1];
            am1.x = Asb[m1 + l16][kidx]; am1.y = Asb[m1 + l16][kidx + 1];
            bn0.x = Bsb[kidx][n0 + l16]; bn0.y = Bsb[kidx + 1][n0 + l16];
            bn1.x = Bsb[kidx][n1 + l16]; bn1.y = Bsb[kidx + 1][n1 + l16];
            acc00 = __builtin_amdgcn_wmma_f32_16x16x4_f32(
                false, am0, false, bn0, (short)0, acc00, false, false);
            acc01 = __builtin_amdgcn_wmma_f32_16x16x4_f32(
                false, am0, false, bn1, (short)0, acc01, false, false);
            acc10 = __builtin_amdgcn_wmma_f32_16x16x4_f32(
                false, am1, false, bn0, (short)0, acc10, false, false);
            acc11 = __builtin_amdgcn_wmma_f32_16x16x4_f32(
                false, am1, false, bn1, (short)0, acc11, false, false);
        }
        __syncthreads();          // all waves done reading buf i
    }

    #pragma unroll
    for (int j = 0; j < 8; ++j) {
        const int r0 = tm + m0 + j + mhalf;
        const int r1 = tm + m1 + j + mhalf;
        Cg[(size_t)r0 * F_ + tn + n0 + l16] = acc00[j];
        Cg[(size_t)r0 * F_ + tn + n1 + l16] = acc01[j];
        Cg[(size_t)r1 * F_ + tn + n0 + l16] = acc10[j];
        Cg[(size_t)r1 * F_ + tn + n1 + l16] = acc11[j];
    }
}

// ---------------------------------------------------------------------------
// Kernel 2: Gram matrices  C[b,m,n] = sum_f X[b,m,f] * Y[b,n,f]
// blockIdx.z selects: 0 FtF(f,f)  1 FtG(f,g)  2 GtG(g,g)  3 GtF(g,f)
// ---------------------------------------------------------------------------
__global__ __launch_bounds__(256) void gram_gemm(
    const float* __restrict__ fhat, const float* __restrict__ ghat,
    float* __restrict__ grams)
{
    __shared__ float As[32][65];   // [f][m]
    __shared__ float Bs[32][65];   // [f][n]

    const int w  = blockIdx.z;
    const int b  = blockIdx.y;
    const int tm = (blockIdx.x >> 1) * 64;
    const int tn = (blockIdx.x & 1)  * 64;
    const int t    = threadIdx.x;
    const int lane = t & 31;
    const int wave = t >> 5;
    const int l16  = lane & 15;
    const int khalf = (lane >> 4) << 1;
    const int mhalf = (lane >> 4) * 8;

    const int s0 = wave * 2, s1 = wave * 2 + 1;
    const int sm0 = (s0 >> 2) * 16, sn0 = (s0 & 3) * 16;
    const int sm1 = (s1 >> 2) * 16, sn1 = (s1 & 3) * 16;

    const float* X = (w < 2) ? fhat : ghat;
    const float* Y = (w == 0 || w == 3) ? fhat : ghat;
    const float* Xg = X + (size_t)b * K_ * F_;
    const float* Yg = Y + (size_t)b * K_ * F_;

    v8f acc0 = {}; v8f acc1 = {};

    for (int f0 = 0; f0 < F_; f0 += 32) {
        const int fi = t & 31, ri0 = t >> 5;
        #pragma unroll
        for (int r = 0; r < 8; ++r) {
            const int ri = ri0 + r * 8;
            As[fi][ri] = Xg[(size_t)(tm + ri) * F_ + f0 + fi];
            Bs[fi][ri] = Yg[(size_t)(tn + ri) * F_ + f0 + fi];
        }
        __syncthreads();

        #pragma unroll
        for (int kk = 0; kk < 32; kk += 4) {
            const int kidx = kk + khalf;
            v2f a0, a1, b0, b1;
            a0.x = As[kidx][sm0 + l16]; a0.y = As[kidx + 1][sm0 + l16];
            b0.x = Bs[kidx][sn0 + l16]; b0.y = Bs[kidx + 1][sn0 + l16];
            a1.x = As[kidx][sm1 + l16]; a1.y = As[kidx + 1][sm1 + l16];
            b1.x = Bs[kidx][sn1 + l16]; b1.y = Bs[kidx + 1][sn1 + l16];
            acc0 = __builtin_amdgcn_wmma_f32_16x16x4_f32(
                false, a0, false, b0, (short)0, acc0, false, false);
            acc1 = __builtin_amdgcn_wmma_f32_16x16x4_f32(
                false, a1, false, b1, (short)0, acc1, false, false);
        }
        __syncthreads();
    }

    float* Cg = grams + (size_t)w * B_ * K_ * K_ + (size_t)b * K_ * K_;
    #pragma unroll
    for (int j = 0; j < 8; ++j) {
        Cg[(size_t)(tm + sm0 + j + mhalf) * K_ + tn + sn0 + l16] = acc0[j];
        Cg[(size_t)(tm + sm1 + j + mhalf) * K_ + tn + sn1 + l16] = acc1[j];
    }
}

// ---------------------------------------------------------------------------
// Kernel 3: in-place Gauss-Jordan solve (SPD Gram matrix => no pivoting)
// of A X = Bm per (batch, pair); writes out = X^T.
// p=0: A=FtF, Bm=FtG -> c_xy ; p=1: A=GtG, Bm=GtF -> c_yx
// ---------------------------------------------------------------------------
__global__ __launch_bounds__(256) void solve_gj(
    float* __restrict__ grams, float* __restrict__ outp)
{
    const int p = blockIdx.x;
    const int b = blockIdx.y;
    const size_t KK = (size_t)K_ * K_;
    const size_t BKK = (size_t)B_ * KK;

    float* A  = grams + (p ? 2 * BKK : 0)   + (size_t)b * KK;
    float* Bm = grams + (p ? 3 * BKK : BKK) + (size_t)b * KK;
    float* Og = outp + (size_t)p * BKK + (size_t)b * KK;

    const int t = threadIdx.x;
    __shared__ float pivRow[256];
    __shared__ float factors[128];
    __shared__ float s_inv;

    for (int c = 0; c < K_; ++c) {
        if (t == 0) s_inv = 1.0f / A[c * K_ + c];
        if (t < K_) factors[t] = A[t * K_ + c];
        __syncthreads();

        float val = (t < K_) ? A[c * K_ + t] : Bm[c * K_ + (t - K_)];
        val *= s_inv;
        pivRow[t] = val;
        if (t < K_) A[c * K_ + t] = val;
        else        Bm[c * K_ + (t - K_)] = val;
        __syncthreads();

        for (int idx = t; idx < K_ * 256; idx += 256) {
            const int r = idx >> 8;
            const int j = idx & 255;
            if (r == c) continue;
            const float f = factors[r];
            if (j < K_) A[r * K_ + j]         -= f * pivRow[j];
            else        Bm[r * K_ + (j - K_)] -= f * pivRow[j];
        }
        __syncthreads();
    }

    for (int idx = t; idx < (int)KK; idx += 256) {
        const int i = idx >> 7;
        const int j = idx & 127;
        Og[idx] = Bm[(size_t)j * K_ + i];
    }
}

// ---------------------------------------------------------------------------
extern "C" void kernel_launch(void* const* d_in, const int* in_sizes, int n_in,
                              void* d_out, int out_size, void* d_ws, size_t ws_size,
                              hipStream_t stream) {
    (void)in_sizes; (void)n_in; (void)out_size; (void)ws_size;

    const float* feat_x = (const float*)d_in[0];
    const float* feat_y = (const float*)d_in[1];
    const float* evx    = (const float*)d_in[2];
    const float* evy    = (const float*)d_in[3];
    float* out = (float*)d_out;

    float* ws    = (float*)d_ws;
    float* f_hat = ws;                                   // B*K*F
    float* g_hat = f_hat + (size_t)B_ * K_ * F_;         // B*K*F
    float* grams = g_hat + (size_t)B_ * K_ * F_;         // 4 * B*K*K

    dim3 blk(256);
    proj_gemm<<<dim3(4, B_, 2), blk, 0, stream>>>(evx, feat_x, evy, feat_y,
                                                  f_hat, g_hat);
    gram_gemm<<<dim3(4, B_, 4), blk, 0, stream>>>(f_hat, g_hat, grams);
    solve_gj <<<dim3(2, B_),    blk, 0, stream>>>(grams, out);
}